// HeteroEmbeddingLayer_17085379903649
// MI455X (gfx1250) — compile-verified
//
#include <hip/hip_runtime.h>
#include <math.h>

typedef __attribute__((ext_vector_type(2))) float v2f;
typedef __attribute__((ext_vector_type(8))) float v8f;

#define N_NODES 50000
#define E_EDGES 800000
#define DRAW    768
#define DF      64
#define NEG_SLOPE 0.2f

// =====================================================================
// GEMM: C[M,64] = A[M,K] @ W[K,64] (+bias), K % 64 == 0, M % 16 == 0.
// Block = 128 threads (4 waves). Wave w computes the 16x16 tile covering
// columns [16w, 16w+16). K staged through LDS in 64-wide chunks; each chunk
// feeds 16 x v_wmma_f32_16x16x4_f32.
// =====================================================================
__global__ __launch_bounds__(128) void gemm_wmma_f32(
    const float* __restrict__ A, const float* __restrict__ W,
    const float* __restrict__ bias, float* __restrict__ C,
    int M, int K, int useBias)
{
  __shared__ __align__(16) float lA[16][68];   // +4 pad: kills bank conflicts on column reads
  __shared__ __align__(16) float lB[64][64];
  const int t    = threadIdx.x;
  const int wave = t >> 5;
  const int lane = t & 31;
  const int m0   = blockIdx.x * 16;

  v8f acc = {};

  for (int k0 = 0; k0 < K; k0 += 64) {
    // cooperative float4 loads: A tile 16x64 (256 float4), B tile 64x64 (1024 float4)
    for (int i = t; i < 16 * 16; i += 128) {
      int r = i >> 4, c = (i & 15) << 2;
      *reinterpret_cast<float4*>(&lA[r][c]) =
          *reinterpret_cast<const float4*>(&A[(size_t)(m0 + r) * K + k0 + c]);
    }
    for (int i = t; i < 64 * 16; i += 128) {
      int r = i >> 4, c = (i & 15) << 2;
      *reinterpret_cast<float4*>(&lB[r][c]) =
          *reinterpret_cast<const float4*>(&W[(size_t)(k0 + r) * DF + c]);
    }
    __syncthreads();

    const int mRow = lane & 15;                 // A row this lane carries
    const int nCol = wave * 16 + (lane & 15);   // B/C column this lane carries
    #pragma unroll
    for (int k4 = 0; k4 < 64; k4 += 4) {
      // ISA layout: lanes 0-15 hold K = k4+0,k4+1 ; lanes 16-31 hold K = k4+2,k4+3
      const int kk = k4 + ((lane >> 4) << 1);   // branch-free: EXEC stays all-1s
      v2f a, b;
      a[0] = lA[mRow][kk + 0];
      a[1] = lA[mRow][kk + 1];
      b[0] = lB[kk + 0][nCol];
      b[1] = lB[kk + 1][nCol];
      acc = __builtin_amdgcn_wmma_f32_16x16x4_f32(
          /*neg_a=*/false, a, /*neg_b=*/false, b,
          /*c_mod=*/(short)0, acc, /*reuse_a=*/false, /*reuse_b=*/false);
    }
    __syncthreads();
  }

  // C layout: VGPR r -> row r (lanes 0-15) / row r+8 (lanes 16-31), col = lane%16
  const int col = wave * 16 + (lane & 15);
  const float bv = useBias ? bias[col] : 0.0f;
  const int rbase = (lane >> 4) << 3;           // 0 or 8
  #pragma unroll
  for (int r = 0; r < 8; ++r) {
    C[(size_t)(m0 + rbase + r) * DF + col] = acc[r] + bv;
  }
}

// ---------------------------------------------------------------------
// el[i] = z[i,:] . a_l ; er[i] = z[i,:] . a_r   (attention projections)
// ---------------------------------------------------------------------
__global__ __launch_bounds__(256) void rowdot_kernel(
    const float* __restrict__ z, const float* __restrict__ al,
    const float* __restrict__ ar, float* __restrict__ el,
    float* __restrict__ er, int n)
{
  __shared__ float s_al[DF], s_ar[DF];
  if (threadIdx.x < DF) { s_al[threadIdx.x] = al[threadIdx.x]; s_ar[threadIdx.x] = ar[threadIdx.x]; }
  __syncthreads();
  const int row = blockIdx.x * blockDim.x + threadIdx.x;
  if (row >= n) return;
  const float* zr = z + (size_t)row * DF;
  float accl = 0.f, accr = 0.f;
  #pragma unroll
  for (int j = 0; j < DF; j += 4) {
    float4 v = *reinterpret_cast<const float4*>(zr + j);
    accl += v.x * s_al[j] + v.y * s_al[j + 1] + v.z * s_al[j + 2] + v.w * s_al[j + 3];
    accr += v.x * s_ar[j] + v.y * s_ar[j + 1] + v.z * s_ar[j + 2] + v.w * s_ar[j + 3];
  }
  el[row] = accl;
  er[row] = accr;
}

// ---------------------------------------------------------------------
// init: rst = 0, m = -inf, s = 0
// ---------------------------------------------------------------------
__global__ void init_kernel(float* __restrict__ rst, float* __restrict__ m,
                            float* __restrict__ s, int n_nodes)
{
  const int i = blockIdx.x * blockDim.x + threadIdx.x;
  if (i < n_nodes * DF) rst[i] = 0.0f;
  if (i < n_nodes) { m[i] = -INFINITY; s[i] = 0.0f; }
}

// ---------------------------------------------------------------------
// pass 1: e = leaky_relu(el[src] + er[dst]); segment max via f32 atomicMax
// ---------------------------------------------------------------------
__global__ void edge_max_kernel(const float* __restrict__ el, const float* __restrict__ er,
                                const int* __restrict__ src, const int* __restrict__ dst,
                                float* __restrict__ e, float* __restrict__ m, int E)
{
  const int i = blockIdx.x * blockDim.x + threadIdx.x;
  if (i >= E) return;
  float v = el[src[i]] + er[dst[i]];
  v = (v >= 0.f) ? v : NEG_SLOPE * v;
  e[i] = v;
  atomicMax(&m[dst[i]], v);                 // global_atomic_max_num_f32 (L2-resident)
}

// ---------------------------------------------------------------------
// pass 2: ex = exp(e - m[dst]); e <- ex; s[dst] += ex
// ---------------------------------------------------------------------
__global__ void edge_expsum_kernel(float* __restrict__ e, const float* __restrict__ m,
                                   const int* __restrict__ dst, float* __restrict__ s, int E)
{
  const int i = blockIdx.x * blockDim.x + threadIdx.x;
  if (i >= E) return;
  const int d = dst[i];
  const float ex = __expf(e[i] - m[d]);
  e[i] = ex;
  atomicAdd(&s[d], ex);
}

// ---------------------------------------------------------------------
// pass 3: alpha = ex / s[dst]; rst[dst,:] += alpha * z[src,:]
// 16 threads per edge, 4 columns each (float4 gather + 4 f32 atomic adds).
// ---------------------------------------------------------------------
__global__ __launch_bounds__(256) void edge_scatter_kernel(
    const float* __restrict__ e, const float* __restrict__ s,
    const float* __restrict__ z, const int* __restrict__ src,
    const int* __restrict__ dst, float* __restrict__ rst, int E)
{
  const int gid  = blockIdx.x * blockDim.x + threadIdx.x;
  const int edge = gid >> 4;
  const int part = gid & 15;
  if (edge >= E) return;
  // speculative prefetch of a future gather row (global_prefetch_b8)
  if (part == 0 && edge + 256 < E)
    __builtin_prefetch(&z[(size_t)src[edge + 256] * DF], 0, 0);
  const int sN = src[edge], dN = dst[edge];
  const float alpha = e[edge] / s[dN];
  const float4 v = *reinterpret_cast<const float4*>(&z[(size_t)sN * DF + part * 4]);
  float* out = &rst[(size_t)dN * DF + part * 4];
  atomicAdd(out + 0, alpha * v.x);
  atomicAdd(out + 1, alpha * v.y);
  atomicAdd(out + 2, alpha * v.z);
  atomicAdd(out + 3, alpha * v.w);
}

// ---------------------------------------------------------------------
// out = elu(rst + h + b)   (residual + bias + ELU, alpha=1)
// ---------------------------------------------------------------------
__global__ void finalize_kernel(const float* __restrict__ rst, const float* __restrict__ h,
                                const float* __restrict__ b, float* __restrict__ out, int n)
{
  const int i = blockIdx.x * blockDim.x + threadIdx.x;
  if (i >= n * DF) return;
  const float x = rst[i] + h[i] + b[i & (DF - 1)];
  out[i] = (x > 0.f) ? x : (__expf(x) - 1.0f);
}

// =====================================================================
extern "C" void kernel_launch(void* const* d_in, const int* in_sizes, int n_in,
                              void* d_out, int out_size, void* d_ws, size_t ws_size,
                              hipStream_t stream)
{
  const float* img_feat = (const float*)d_in[0];
  const float* txt_feat = (const float*)d_in[1];
  const int*   img_src  = (const int*)d_in[2];
  const int*   img_dst  = (const int*)d_in[3];
  const int*   txt_src  = (const int*)d_in[4];
  const int*   txt_dst  = (const int*)d_in[5];
  const float* W_mm     = (const float*)d_in[6];
  const float* b_mm     = (const float*)d_in[7];
  const float* fc_img   = (const float*)d_in[8];
  const float* al_img   = (const float*)d_in[9];
  const float* ar_img   = (const float*)d_in[10];
  const float* b_img    = (const float*)d_in[11];
  const float* fc_txt   = (const float*)d_in[12];
  const float* al_txt   = (const float*)d_in[13];
  const float* ar_txt   = (const float*)d_in[14];
  const float* b_txt    = (const float*)d_in[15];
  float* out = (float*)d_out;

  // workspace carve-out
  char* ws = (char*)d_ws;
  size_t off = 0;
  auto carve = [&](size_t bytes) -> float* {
    float* p = (float*)(ws + off);
    off += (bytes + 255) & ~(size_t)255;
    return p;
  };
  float* h_img = carve((size_t)N_NODES * DF * 4);
  float* h_txt = carve((size_t)N_NODES * DF * 4);
  float* z     = carve((size_t)N_NODES * DF * 4);
  float* rst   = carve((size_t)N_NODES * DF * 4);
  float* el    = carve((size_t)N_NODES * 4);
  float* er    = carve((size_t)N_NODES * 4);
  float* mbuf  = carve((size_t)N_NODES * 4);
  float* sbuf  = carve((size_t)N_NODES * 4);
  float* ebuf  = carve((size_t)E_EDGES * 4);
  (void)ws_size; (void)in_sizes; (void)n_in; (void)out_size;

  const dim3 gB(128), gG(N_NODES / 16);        // 3125 blocks, M%16==0
  const int nodeElemBlocks = (N_NODES * DF + 255) / 256;
  const int nodeBlocks     = (N_NODES + 255) / 256;
  const int edgeBlocks     = (E_EDGES + 255) / 256;
  const int scatterBlocks  = (E_EDGES * 16 + 255) / 256;

  // multimodal projection: h = feat @ W_mm + b_mm  (K = 768)
  gemm_wmma_f32<<<gG, gB, 0, stream>>>(img_feat, W_mm, b_mm, h_img, N_NODES, DRAW, 1);
  gemm_wmma_f32<<<gG, gB, 0, stream>>>(txt_feat, W_mm, b_mm, h_txt, N_NODES, DRAW, 1);

  for (int l = 0; l < 2; ++l) {
    struct Ty {
      float* h; const int* src; const int* dst;
      const float* fc; const float* al; const float* ar; const float* b;
      float* outp;
    };
    Ty ty[2] = {
      { h_img, img_src, img_dst, fc_img + l * DF * DF, al_img + l * DF, ar_img + l * DF,
        b_img + l * DF, (l == 1) ? out : h_img },
      { h_txt, txt_src, txt_dst, fc_txt + l * DF * DF, al_txt + l * DF, ar_txt + l * DF,
        b_txt + l * DF, (l == 1) ? out + (size_t)N_NODES * DF : h_txt },
    };
    for (int k = 0; k < 2; ++k) {
      // z = h @ W_l   (K = 64)
      gemm_wmma_f32<<<gG, gB, 0, stream>>>(ty[k].h, ty[k].fc, nullptr, z, N_NODES, DF, 0);
      rowdot_kernel<<<nodeBlocks, 256, 0, stream>>>(z, ty[k].al, ty[k].ar, el, er, N_NODES);
      init_kernel<<<nodeElemBlocks, 256, 0, stream>>>(rst, mbuf, sbuf, N_NODES);
      edge_max_kernel<<<edgeBlocks, 256, 0, stream>>>(el, er, ty[k].src, ty[k].dst, ebuf, mbuf, E_EDGES);
      edge_expsum_kernel<<<edgeBlocks, 256, 0, stream>>>(ebuf, mbuf, ty[k].dst, sbuf, E_EDGES);
      edge_scatter_kernel<<<scatterBlocks, 256, 0, stream>>>(ebuf, sbuf, z, ty[k].src, ty[k].dst, rst, E_EDGES);
      finalize_kernel<<<nodeElemBlocks, 256, 0, stream>>>(rst, ty[k].h, ty[k].b, ty[k].outp, N_NODES);
    }
  }
}